// SideChainModel_45655502357211
// MI455X (gfx1250) — compile-verified
//
#include <hip/hip_runtime.h>
#include <math.h>

// ---- problem dims ----
#define NRES  (1024 * 256)   // B*L residues
#define BLK   256            // threads per block (8 wave32)
#define FPT   42             // floats per residue (14 atoms * 3)
#define LDSF  (BLK * FPT)    // 10752 floats = 43008 bytes per block tile

// ---- residue tables ----
__constant__ int c_NSC[21]  = {1,7,4,4,2,5,5,0,6,4,4,5,4,7,3,2,3,10,8,3,0};
__constant__ int c_CHIN[21] = {0,4,2,2,1,3,3,0,2,2,2,4,3,2,2,1,1,2,2,1,0};

// ---- geometry constants (cos/sin folded at compile time) ----
#define BL     1.52f
#define NCA    0.35807335f        // -cos(1.937)
#define SBA    0.93369329f        //  sin(1.937)
#define CT_CB  (-0.53896200f)     //  cos(-2.14)
#define ST_CB  (-0.84233250f)     //  sin(-2.14)
#define CT_PI  (-1.0f)            //  cos(pi_f32)
#define ST_PI  (-8.7422777e-8f)   //  sin(pi_f32)

// ---- Adam constants ----
#define AB1   0.9f
#define AB2   0.999f
#define AEPS  1e-8f
#define DELTA 1e-4f
#define MAXT  200

struct F3 { float x, y, z; };
__device__ __forceinline__ F3 f3(float a, float b, float c) { return {a, b, c}; }
__device__ __forceinline__ F3 sub3(F3 a, F3 b) { return {a.x-b.x, a.y-b.y, a.z-b.z}; }
__device__ __forceinline__ float dot3(F3 a, F3 b) { return a.x*b.x + a.y*b.y + a.z*b.z; }
__device__ __forceinline__ F3 cross3(F3 a, F3 b) {
  return {a.y*b.z - a.z*b.y, a.z*b.x - a.x*b.z, a.x*b.y - a.y*b.x};
}
__device__ __forceinline__ F3 unit3(F3 v) {   // reference: v / (||v|| + 1e-8)
  float inv = 1.0f / (sqrtf(dot3(v, v)) + 1e-8f);
  return {v.x*inv, v.y*inv, v.z*inv};
}

__device__ __forceinline__ float dihed3(F3 a, F3 b, F3 c, F3 d) {
  F3 b1 = sub3(b, a), b2 = sub3(c, b), b3 = sub3(d, c);
  F3 n1 = cross3(b1, b2), n2 = cross3(b2, b3);
  F3 m1 = cross3(n1, unit3(b2));
  return atan2f(dot3(m1, n2), dot3(n1, n2));
}

// NeRF extension; also returns the bond axis (unit(c-b)) for the analytic grad
__device__ __forceinline__ F3 extend3(F3 a, F3 b, F3 c, float ct, float st, F3* axis) {
  F3 bc = unit3(sub3(c, b));
  F3 n  = unit3(cross3(sub3(b, a), bc));
  F3 m  = cross3(n, bc);
  *axis = bc;
  return { c.x + BL * (NCA * bc.x + SBA * (ct * m.x + st * n.x)),
           c.y + BL * (NCA * bc.y + SBA * (ct * m.y + st * n.y)),
           c.z + BL * (NCA * bc.z + SBA * (ct * m.z + st * n.z)) };
}

// Forward rebuild + exact analytic gradient wrt chi[0..3].
// chi_k rigidly rotates all downstream atoms about its bond axis, so
// d(p_j)/d(chi_k) = axis_k x (p_j - pivot_k)  -- no autodiff chain needed.
__device__ __forceinline__ float forward_grad(const F3 ori[14], F3 pos[14],
                                              const float chi[4], int nsc,
                                              float inv_gden, float g[4]) {
  static constexpr int PA[10] = {0,0,1,4,5,6,7,8,9,10};
  static constexpr int PB[10] = {2,1,4,5,6,7,8,9,10,11};
  static constexpr int PC[10] = {1,4,5,6,7,8,9,10,11,12};

  float cs[4], sn[4];
#pragma unroll
  for (int k = 0; k < 4; ++k) __sincosf(chi[k], &sn[k], &cs[k]);

  F3 ax[4], piv[4];
#pragma unroll
  for (int s = 0; s < 10; ++s) {
    float ct, st;
    if (s == 0)      { ct = CT_CB;    st = ST_CB;    }
    else if (s <= 4) { ct = cs[s-1];  st = sn[s-1];  }
    else             { ct = CT_PI;    st = ST_PI;    }
    F3 axis;
    pos[4 + s] = extend3(pos[PA[s]], pos[PB[s]], pos[PC[s]], ct, st, &axis);
    if (s >= 1 && s <= 4) { ax[s-1] = axis; piv[s-1] = pos[PC[s]]; }
  }

  float msum = 0.0f;
  g[0] = g[1] = g[2] = g[3] = 0.0f;
#pragma unroll
  for (int s = 0; s < 10; ++s) {
    float w = (s < nsc) ? 1.0f : 0.0f;
    F3 diff = sub3(pos[4 + s], ori[4 + s]);
    msum += w * dot3(diff, diff);
#pragma unroll
    for (int k = 0; k < 4; ++k) {
      if (s >= k + 1) {   // chi_k only moves atoms placed at slot >= k+1
        F3 r = sub3(pos[4 + s], piv[k]);
        g[k] += w * 2.0f * dot3(diff, cross3(ax[k], r));
      }
    }
  }
#pragma unroll
  for (int k = 0; k < 4; ++k) g[k] *= inv_gden;
  return msum;
}

// ---- TDM descriptor helper (D# group0 + group1, 1-D dword tile) ----
typedef unsigned int u32x4 __attribute__((ext_vector_type(4)));
typedef int          i32x8 __attribute__((ext_vector_type(8)));
typedef int          i32x4 __attribute__((ext_vector_type(4)));

__device__ __forceinline__ u32x4 tdm_g0(unsigned long long gaddr, unsigned lds_addr) {
  u32x4 g;
  g.x = 1u;                                               // count=1, user desc
  g.y = lds_addr;                                         // lds_addr [63:32]
  g.z = (unsigned)(gaddr & 0xffffffffull);                // global_addr lo
  g.w = (unsigned)((gaddr >> 32) & 0x01ffffffull)         // global_addr[56:32]
        | (2u << 30);                                     // type=2 ("image")
  return g;
}
__device__ __forceinline__ i32x8 tdm_g1() {
  i32x8 g;
  g.s0 = (int)(2u << 16);                    // data_size=4B, no mcast/pad/iter
  g.s1 = (int)((LDSF & 0xffffu) << 16);      // tensor_dim0[15:0] at bits 63:48
  g.s2 = (int)(((LDSF >> 16) & 0xffffu)      // tensor_dim0[31:16]
               | (1u << 16));                // tensor_dim1 = 1
  g.s3 = (int)((unsigned)LDSF << 16);        // tile_dim0 = 10752 dwords
  g.s4 = 0;                                  // tile_dim1/2 unused (1-D)
  g.s5 = (int)LDSF;                          // tensor_dim0_stride
  g.s6 = 0;
  g.s7 = 0;
  return g;
}

// ---- pass 1: exact global denom = sum(n_sc) (reference: sc_mask.sum()*3) ----
__global__ __launch_bounds__(256) void denom_kernel(const int* __restrict__ S,
                                                    unsigned* __restrict__ acc) {
  int i = blockIdx.x * 256 + threadIdx.x;
  if (i < NRES) atomicAdd(acc, (unsigned)c_NSC[S[i] % 21]);
}

// ---- pass 2: chi0 -> Adam refit -> rebuild, all in registers; TDM staging ----
__global__ __launch_bounds__(256) void sidechain_kernel(const float* __restrict__ Xg,
                                                        const int* __restrict__ Sg,
                                                        float* __restrict__ outg,
                                                        const unsigned* __restrict__ dacc) {
  __shared__ float lds[LDSF];
  const unsigned tid = threadIdx.x;
  const unsigned blk = blockIdx.x;

  // Stage this block's 42KB coordinate tile into LDS via the Tensor Data Mover.
  if (tid < 32) {   // one wave issues the DMA (TDM ignores EXEC; 1 issue/wave)
    unsigned long long ga = (unsigned long long)(const void*)Xg
                          + (unsigned long long)blk * (unsigned long long)(LDSF * 4);
    i32x4 gz4 = {0, 0, 0, 0};
    i32x8 gz8 = {0, 0, 0, 0, 0, 0, 0, 0};
    __builtin_amdgcn_tensor_load_to_lds(tdm_g0(ga, 0u), tdm_g1(), gz4, gz4, gz8, 0);
    __builtin_amdgcn_s_wait_tensorcnt(0);
  }
  __syncthreads();

  // Per-residue working set -> registers.
  const int r = (int)(blk * BLK + tid);
  F3 ori[14];
  {
    const float* mine = &lds[tid * FPT];
#pragma unroll
    for (int i = 0; i < 14; ++i) ori[i] = f3(mine[3*i], mine[3*i+1], mine[3*i+2]);
  }
  const int s20 = Sg[r] % 21;                 // AA_MAP = arange(26) % 21
  const int nsc = c_NSC[s20];
  const int cn  = c_CHIN[s20];
  const float inv_gden = 1.0f / ((float)(*dacc) * 3.0f);            // exact global denom
  const float linv = (nsc > 0) ? 1.0f / (3.0f * (float)nsc) : 0.0f; // per-res normalizer

  // chi0 = masked dihedrals over CHI_QUADS
  static constexpr int QA[4] = {0,1,4,5}, QB[4] = {1,4,5,6},
                       QC[4] = {4,5,6,7}, QD[4] = {5,6,7,8};
  float chi[4];
#pragma unroll
  for (int k = 0; k < 4; ++k)
    chi[k] = (k < cn) ? dihed3(ori[QA[k]], ori[QB[k]], ori[QC[k]], ori[QD[k]]) : 0.0f;

  F3 pos[14];
#pragma unroll
  for (int i = 0; i < 4; ++i) pos[i] = ori[i];   // backbone fixed

  // Adam refit (reference semantics; stop test on per-residue normalized MSE)
  float m[4] = {0,0,0,0}, v[4] = {0,0,0,0}, g[4];
  float msum = forward_grad(ori, pos, chi, nsc, inv_gden, g);
  float mse = msum * linv, last = 100.0f;
  float b1t = 1.0f, b2t = 1.0f;
#pragma unroll 1
  for (int t = 0; t < MAXT; ++t) {
    if (fabsf(mse - last) < DELTA) break;
    b1t *= AB1; b2t *= AB2;
#pragma unroll
    for (int k = 0; k < 4; ++k) {
      m[k] = AB1 * m[k] + (1.0f - AB1) * g[k];
      v[k] = AB2 * v[k] + (1.0f - AB2) * g[k] * g[k];
      float mh = m[k] / (1.0f - b1t);
      float vh = v[k] / (1.0f - b2t);
      chi[k] -= mh / (sqrtf(vh) + AEPS);
    }
    last = mse;
    msum = forward_grad(ori, pos, chi, nsc, inv_gden, g);
    mse = msum * linv;
  }
  // pos[] now holds build_sidechain(chi_final); emit backbone + all 10 rebuilt atoms.
  {
    float* mine = &lds[tid * FPT];
#pragma unroll
    for (int i = 0; i < 4; ++i)  { mine[3*i] = ori[i].x; mine[3*i+1] = ori[i].y; mine[3*i+2] = ori[i].z; }
#pragma unroll
    for (int i = 4; i < 14; ++i) { mine[3*i] = pos[i].x; mine[3*i+1] = pos[i].y; mine[3*i+2] = pos[i].z; }
  }
  __syncthreads();

  // Stream the finished tile back out through the TDM.
  if (tid < 32) {
    unsigned long long ga = (unsigned long long)(void*)outg
                          + (unsigned long long)blk * (unsigned long long)(LDSF * 4);
    i32x4 gz4 = {0, 0, 0, 0};
    i32x8 gz8 = {0, 0, 0, 0, 0, 0, 0, 0};
    __builtin_amdgcn_tensor_store_from_lds(tdm_g0(ga, 0u), tdm_g1(), gz4, gz4, gz8, 0);
    __builtin_amdgcn_s_wait_tensorcnt(0);
  }
}

extern "C" void kernel_launch(void* const* d_in, const int* in_sizes, int n_in,
                              void* d_out, int out_size, void* d_ws, size_t ws_size,
                              hipStream_t stream) {
  const float* X = (const float*)d_in[0];
  const int*   S = (const int*)d_in[1];
  // d_in[2] (batch_ids) is B contiguous equal-length segments -> pure reshape, unused.
  float* out = (float*)d_out;
  unsigned* acc = (unsigned*)d_ws;

  (void)hipMemsetAsync(acc, 0, sizeof(unsigned), stream);
  hipLaunchKernelGGL(denom_kernel, dim3(NRES / BLK), dim3(BLK), 0, stream, S, acc);
  hipLaunchKernelGGL(sidechain_kernel, dim3(NRES / BLK), dim3(BLK), 0, stream,
                     X, S, out, acc);
}